// LFPointTransformer_61546881352057
// MI455X (gfx1250) — compile-verified
//
#include <hip/hip_runtime.h>
#include <math.h>

typedef __attribute__((ext_vector_type(2))) float v2f;
typedef __attribute__((ext_vector_type(8))) float v8f;

static constexpr int kB      = 32;
static constexpr int kN      = 2000;
static constexpr int kD      = 5;
static constexpr int kDP     = 8;    // D padded to 8 (two K=4 WMMA steps)
static constexpr int kGroup  = 5;
static constexpr int kNG     = kN / kGroup;   // 400 groups
static constexpr int kTopK   = 110;
static constexpr int kNTiles = kN / 16;       // 125 (exact)

// ---------------------------------------------------------------------------
// Kernel 1: x = in@Wi+bi ; Q = (x@Wq+bq)/sqrt(5) ; K = x@Wk+bk
// Writes padded [B,N,8] rows (cols 5..7 = 0) for WMMA consumption.
// ---------------------------------------------------------------------------
__global__ void proj_kernel(const float* __restrict__ in_mat,
                            const float* __restrict__ Wi, const float* __restrict__ bi,
                            const float* __restrict__ Wq, const float* __restrict__ bq,
                            const float* __restrict__ Wk, const float* __restrict__ bk,
                            float* __restrict__ Qp, float* __restrict__ Kp) {
    int idx = blockIdx.x * blockDim.x + threadIdx.x;
    if (idx >= kB * kN) return;
    const float* p = in_mat + (size_t)idx * 3;
    float i0 = p[0], i1 = p[1], i2 = p[2];
    float x[kD];
#pragma unroll
    for (int j = 0; j < kD; ++j)
        x[j] = bi[j] + i0 * Wi[0 * kD + j] + i1 * Wi[1 * kD + j] + i2 * Wi[2 * kD + j];
    const float scale = 0.44721359549995793f; // 1/sqrt(5), folded into Q
    float* qr = Qp + (size_t)idx * kDP;
    float* kr = Kp + (size_t)idx * kDP;
#pragma unroll
    for (int j = 0; j < kD; ++j) {
        float q = bq[j], k = bk[j];
#pragma unroll
        for (int c = 0; c < kD; ++c) { q += x[c] * Wq[c * kD + j]; k += x[c] * Wk[c * kD + j]; }
        qr[j] = q * scale;
        kr[j] = k;
    }
    qr[5] = 0.f; qr[6] = 0.f; qr[7] = 0.f;
    kr[5] = 0.f; kr[6] = 0.f; kr[7] = 0.f;
}

// ---------------------------------------------------------------------------
// Kernel 2: diagonal of softmax(Q K^T) per (batch, 16-row n-tile).
// Scores are bounded tiny (|s| << 88) so softmax is computed shift-free:
// diag = exp(s_nn) / sum_m exp(s_nm) — identical to the max-subtracted form.
// 8 waves/block stream m-tiles through WMMA accumulating per-lane sum-exp;
// wave 0 recomputes the single diagonal tile once (branch-free extract).
//
// Layouts (CDNA5 ISA 7.12.2, 32-bit, wave32):
//   A 16x4: lane&15 = row M, lane>>4 selects K pair {2h, 2h+1} across 2 VGPRs
//   B 4x16: lane&15 = col N, lane>>4 selects K pair (symmetric to A)
//   C 16x16: VGPR v, lane l -> row v + 8*(l>>4), col l&15
// ---------------------------------------------------------------------------
__global__ __launch_bounds__(256) void diag_softmax_kernel(const float* __restrict__ Qp,
                                                           const float* __restrict__ Kp,
                                                           float* __restrict__ diag_out) {
    const int t    = blockIdx.x;            // n-tile index (0..124)
    const int b    = blockIdx.y;            // batch
    const int tid  = threadIdx.x;
    const int lane = tid & 31;
    const int wave = tid >> 5;
    const int h    = lane >> 4;             // half-wave
    const int col  = lane & 15;
    const int n0   = t * 16;

    __shared__ float ssum[8 * 32 * 8];
    __shared__ float sdiag[16];

    const float* Qb = Qp + (size_t)b * kN * kDP;
    const float* Kb = Kp + (size_t)b * kN * kDP;

    // A-matrix (Q rows for this n-tile) is loop-invariant.
    const float* qrow = Qb + (size_t)(n0 + col) * kDP;
    v2f a_lo = *(const v2f*)(qrow + 2 * h);       // K = 0..3
    v2f a_hi = *(const v2f*)(qrow + 4 + 2 * h);   // K = 4..7 (5..7 are zero pad)

    float rsum[8];
#pragma unroll
    for (int v = 0; v < 8; ++v) rsum[v] = 0.f;

    for (int mt = wave; mt < kNTiles; mt += 8) {   // uniform per wave -> EXEC all-1s at WMMA
        const float* krow = Kb + (size_t)(mt * 16 + col) * kDP;
        v2f b_lo = *(const v2f*)(krow + 2 * h);
        v2f b_hi = *(const v2f*)(krow + 4 + 2 * h);

        v8f c = {};
        c = __builtin_amdgcn_wmma_f32_16x16x4_f32(false, a_lo, false, b_lo, (short)0, c, false, false);
        c = __builtin_amdgcn_wmma_f32_16x16x4_f32(false, a_hi, false, b_hi, (short)0, c, false, false);

#pragma unroll
        for (int v = 0; v < 8; ++v)   // row r = v + 8h, this lane's column
            rsum[v] += __expf(c[v]);
    }

#pragma unroll
    for (int v = 0; v < 8; ++v)
        ssum[(wave * 32 + lane) * 8 + v] = rsum[v];

    if (wave == 0) {  // recompute the diagonal tile once; extract s_nn branch-free
        const float* krow = Kb + (size_t)(n0 + col) * kDP;
        v2f b_lo = *(const v2f*)(krow + 2 * h);
        v2f b_hi = *(const v2f*)(krow + 4 + 2 * h);
        v8f c = {};
        c = __builtin_amdgcn_wmma_f32_16x16x4_f32(false, a_lo, false, b_lo, (short)0, c, false, false);
        c = __builtin_amdgcn_wmma_f32_16x16x4_f32(false, a_hi, false, b_hi, (short)0, c, false, false);
        // diag element (r == col): r = v + 8h -> lane holds it iff (col>>3)==h, at v = col&7
        int vsel = col & 7;
        float dval = c[0];
#pragma unroll
        for (int v = 1; v < 8; ++v) dval = (vsel == v) ? c[v] : dval;   // cndmask chain
        if ((col >> 3) == h) sdiag[col] = dval;                          // one predicated store
    }
    __syncthreads();

    if (tid < 16) {  // one thread per row r: reduce 8 waves x 16 lanes of partial sums
        int r = tid, hh = r >> 3, v = r & 7;
        float S = 0.f;
        for (int w = 0; w < 8; ++w) {
#pragma unroll
            for (int l = 0; l < 16; ++l)
                S += ssum[(w * 32 + (l + 16 * hh)) * 8 + v];
        }
        diag_out[(size_t)b * kN + n0 + r] = __expf(sdiag[r]) / S;
    }
}

// ---------------------------------------------------------------------------
// Kernel 3: per batch — group-of-5 argmax on diag, centroid, distances,
// iterative top-110 selection (ascending distance, ties -> lower index),
// gather selected points to output. One block per batch.
// ---------------------------------------------------------------------------
__global__ __launch_bounds__(256) void select_kernel(const float* __restrict__ in_mat,
                                                     const float* __restrict__ diag,
                                                     float* __restrict__ out) {
    const int b = blockIdx.x;
    const int t = threadIdx.x;
    __shared__ float dist[kN];
    __shared__ float rv[256];
    __shared__ int   ri[256];
    __shared__ float cent[3];

    // --- group sums + argmax (first-max tie-break, like jnp.argmax) ---
    float bv = -3.0e38f; int bg = 0x7fffffff;
    for (int g = t; g < kNG; g += 256) {
        const float* dp = diag + (size_t)b * kN + g * kGroup;
        float s = dp[0] + dp[1] + dp[2] + dp[3] + dp[4];
        if (s > bv) { bv = s; bg = g; }   // strided ascending g: strict > keeps lowest idx
    }
    rv[t] = bv; ri[t] = bg;
    __syncthreads();
    for (int st = 128; st > 0; st >>= 1) {
        if (t < st) {
            float ov = rv[t + st]; int oi = ri[t + st];
            if (ov > rv[t] || (ov == rv[t] && oi < ri[t])) { rv[t] = ov; ri[t] = oi; }
        }
        __syncthreads();
    }
    int gbest = ri[0];

    if (t < 3) {  // centroid of the 5 winning points, per coordinate
        float s = 0.f;
        for (int p = 0; p < kGroup; ++p)
            s += in_mat[((size_t)b * kN + gbest * kGroup + p) * 3 + t];
        cent[t] = s * 0.2f;
    }
    __syncthreads();

    for (int n = t; n < kN; n += 256) {
        const float* pp = in_mat + ((size_t)b * kN + n) * 3;
        float dx = pp[0] - cent[0], dy = pp[1] - cent[1], dz = pp[2] - cent[2];
        dist[n] = sqrtf(dx * dx + dy * dy + dz * dz);
    }
    __syncthreads();

    for (int it = 0; it < kTopK; ++it) {
        float mv = 3.4e38f; int mi = 0x7fffffff;
        for (int n = t; n < kN; n += 256) {
            float dv = dist[n];
            if (dv < mv) { mv = dv; mi = n; }
        }
        rv[t] = mv; ri[t] = mi;
        __syncthreads();
        for (int st = 128; st > 0; st >>= 1) {
            if (t < st) {
                float ov = rv[t + st]; int oi = ri[t + st];
                if (ov < rv[t] || (ov == rv[t] && oi < ri[t])) { rv[t] = ov; ri[t] = oi; }
            }
            __syncthreads();
        }
        int sel = ri[0];
        if (t < 3) out[((size_t)b * kTopK + it) * 3 + t] =
                       in_mat[((size_t)b * kN + sel) * 3 + t];
        if (t == 0) dist[sel] = 3.4e38f;
        __syncthreads();
    }
}

// ---------------------------------------------------------------------------
extern "C" void kernel_launch(void* const* d_in, const int* in_sizes, int n_in,
                              void* d_out, int out_size, void* d_ws, size_t ws_size,
                              hipStream_t stream) {
    (void)in_sizes; (void)n_in; (void)out_size; (void)ws_size;
    const float* in_mat = (const float*)d_in[0];
    const float* Wi = (const float*)d_in[1];
    const float* bi = (const float*)d_in[2];
    const float* Wq = (const float*)d_in[3];
    const float* bq = (const float*)d_in[4];
    const float* Wk = (const float*)d_in[5];
    const float* bk = (const float*)d_in[6];
    // d_in[7..10] (Wv,bv,Wo,bo) are dead code in the reference -> unused.
    float* out = (float*)d_out;

    // Workspace: Qp[B*N*8] | Kp[B*N*8] | diag[B*N]  (~4.35 MB, fully rewritten)
    float* Qp = (float*)d_ws;
    float* Kp = Qp + (size_t)kB * kN * kDP;
    float* dg = Kp + (size_t)kB * kN * kDP;

    proj_kernel<<<(kB * kN + 255) / 256, 256, 0, stream>>>(in_mat, Wi, bi, Wq, bq, Wk, bk, Qp, Kp);
    dim3 g2(kNTiles, kB);
    diag_softmax_kernel<<<g2, 256, 0, stream>>>(Qp, Kp, dg);
    select_kernel<<<kB, 256, 0, stream>>>(in_mat, dg, out);
}